// NetVLAD_7859790151904
// MI455X (gfx1250) — compile-verified
//
#include <hip/hip_runtime.h>
#include <hip/hip_bf16.h>
#include <math.h>

// NetVLAD fused for gfx1250 (MI455X): wave32, V_WMMA_F32_16X16X4_F32 for both
// GEMMs, TENSOR_LOAD_TO_LDS (TDM) double-buffered HBM->LDS staging.
// Sizes: N=32, C=128, S=16384, K=64. HBM-bound: x (256 MiB) read exactly once.

typedef __attribute__((ext_vector_type(2))) float v2f;
typedef __attribute__((ext_vector_type(8))) float v8f;
typedef unsigned int v4u __attribute__((ext_vector_type(4)));
typedef int v8i_t __attribute__((ext_vector_type(8)));
typedef int v4i_t __attribute__((ext_vector_type(4)));

#define NN 32
#define CC 128
#define SS 16384
#define KK 64
#define SB 64            // s-positions per chunk (TDM tile width)
#define CHUNKS 16        // chunks per workgroup
#define SBLK (SB*CHUNKS) // 1024
#define NSB (SS/SBLK)    // 16 s-blocks per image
#define KCOLS 144        // 128 channels + ones-col + 15 zero pad (9 N-tiles)
#define XS 65            // x tile LDS row stride (odd, matches TDM pad)
#define WS 129           // conv weight LDS stride (odd)
#define AS 65            // assignment LDS stride (odd)

static __device__ __forceinline__ v8f wmma_f32_4(v2f a, v2f b, v8f c) {
  return __builtin_amdgcn_wmma_f32_16x16x4_f32(
      false, a, false, b, (short)0, c, false, false);
}

// One TDM op: DMA a [c=128][s=64] f32 tile (row stride S in memory) into LDS
// at lds_byte_off with row stride 65 DWORDs (pad 1 DWORD every 64).
static __device__ __forceinline__ void tdm_load_chunk(const float* gsrc,
                                                      unsigned lds_byte_off) {
  unsigned long long ga = (unsigned long long)(size_t)gsrc;
  v4u g0;
  g0.x = 0x1u;                                        // count=1, user mode
  g0.y = lds_byte_off;                                // LDS destination
  g0.z = (unsigned)(ga & 0xFFFFFFFFu);                // global_addr[31:0]
  g0.w = (unsigned)((ga >> 32) & 0x01FFFFFFu)         // global_addr[56:32]
         | 0x80000000u;                               // type=2 ("image")
  v8i_t g1;
  g1[0] = (int)((2u << 16)       // data_size = 4B
                | (1u << 20)     // pad_enable
                | (5u << 22));   // pad_interval: every 64 DWORDs (+1 DW pad)
  g1[1] = (int)(16384u << 16);            // tensor_dim0[15:0] = S
  g1[2] = (int)((16384u >> 16) | (128u << 16)); // dim0 hi | tensor_dim1 = C
  g1[3] = (int)(64u << 16);               // tensor_dim1 hi | tile_dim0 = 64
  g1[4] = (int)(128u);                    // tile_dim1 = 128, tile_dim2 = 0
  g1[5] = (int)(16384u);                  // tensor_dim0_stride lo = S
  g1[6] = 0;                              // stride hi | dim1_stride lo
  g1[7] = 0;
  v4i_t g2 = {0, 0, 0, 0};
  v4i_t g3 = {0, 0, 0, 0};
  v8i_t g4 = {0, 0, 0, 0, 0, 0, 0, 0};    // unused trailing group (6-arg form)
  __builtin_amdgcn_tensor_load_to_lds(g0, g1, g2, g3, g4, 0);
}

__global__ __launch_bounds__(256, 2)
void netvlad_main(const float* __restrict__ x,      // [N][C][S]
                  const float* __restrict__ convw,  // [K][C]
                  float* __restrict__ partial)      // [N*NSB][64][144]
{
  // Double-buffered x tile: rows 0..127 = channels (TDM-written),
  // rows 128..143 constant: row 128 = ones (sum_a column), 129..143 = 0.
  __shared__ float xb[2][KCOLS * XS];   // 2 * 37440 B
  __shared__ float wl[KK * WS];         // 33024 B
  __shared__ float al[KK * AS];         // 16640 B

  const int tid  = threadIdx.x;
  const int lane = tid & 31;
  const int wave = tid >> 5;
  const int wg   = blockIdx.x;
  const int n    = wg / NSB;
  const int sb   = wg % NSB;
  const int s0   = sb * SBLK;

  const int m  = lane & 15;         // within-16 index (A:M, B:N, D:N)
  const int kb = (lane >> 4) * 2;   // K-pair selected by half-wave
  const float* xn = x + (size_t)n * CC * SS + s0;

  // Kick off TDM for chunk 0 into buffer 0 (wave 0 owns the TDM queue).
  if (wave == 0) tdm_load_chunk(xn, (unsigned)(size_t)&xb[0][0]);

  // Stage conv weights (padded stride).
  for (int i = tid; i < KK * CC; i += 256)
    wl[(i >> 7) * WS + (i & 127)] = convw[i];
  // Constant augment rows in BOTH buffers (written once).
  for (int i = tid; i < 2 * 16 * SB; i += 256) {
    int b = i / (16 * SB);
    int r = (i / SB) & 15;
    int s = i & (SB - 1);
    xb[b][(128 + r) * XS + s] = (r == 0) ? 1.0f : 0.0f;
  }

  // GEMM2 accumulators persist across chunks: 36 tiles (4 M x 9 N) / 8 waves.
  v8f vacc[5];
#pragma unroll
  for (int t = 0; t < 5; ++t) vacc[t] = (v8f){0.f,0.f,0.f,0.f,0.f,0.f,0.f,0.f};

  for (int ch = 0; ch < CHUNKS; ++ch) {
    if (wave == 0) __builtin_amdgcn_s_wait_tensorcnt(0);  // chunk ch landed
    __syncthreads();                                      // publish TDM writes
    // Prefetch chunk ch+1 into the other buffer; overlaps all compute below.
    if (wave == 0 && ch + 1 < CHUNKS)
      tdm_load_chunk(xn + (size_t)(ch + 1) * SB,
                     (unsigned)(size_t)&xb[(ch + 1) & 1][0]);

    const float* xc = &xb[ch & 1][0];

    // ---- GEMM1: logits[64 x 64] = W[64 x 128] * x[128 x 64]
    {
      const int mt  = wave & 3;
      const int ntb = (wave >> 2) * 2;
      v8f l0 = (v8f){0,0,0,0,0,0,0,0}, l1 = l0;
      const int arow = (16 * mt + m) * WS;
      const int b0c = 16 * ntb + m, b1c = 16 * (ntb + 1) + m;
      for (int c = 0; c < CC; c += 4) {
        v2f af;
        af.x = wl[arow + c + kb];
        af.y = wl[arow + c + kb + 1];
        v2f bf0;
        bf0.x = xc[(c + kb) * XS + b0c];
        bf0.y = xc[(c + kb + 1) * XS + b0c];
        l0 = wmma_f32_4(af, bf0, l0);
        v2f bf1;
        bf1.x = xc[(c + kb) * XS + b1c];
        bf1.y = xc[(c + kb + 1) * XS + b1c];
        l1 = wmma_f32_4(af, bf1, l1);
      }
#pragma unroll
      for (int r = 0; r < 8; ++r) {
        const int row = (16 * mt + r + 8 * (lane >> 4)) * AS;
        al[row + 16 * ntb + m] = l0[r];
        al[row + 16 * (ntb + 1) + m] = l1[r];
      }
    }
    __syncthreads();

    // ---- Softmax over k per column: 8 columns/wave, 4 lanes/column.
    {
      const int s  = 8 * wave + (lane & 7);
      const int kq = lane >> 3;           // 0..3 -> 16 k's each
      const int kbase = kq * 16;
      float mx = -3.402823466e38f;
#pragma unroll
      for (int i = 0; i < 16; ++i) mx = fmaxf(mx, al[(kbase + i) * AS + s]);
      mx = fmaxf(mx, __shfl_xor(mx, 8, 32));
      mx = fmaxf(mx, __shfl_xor(mx, 16, 32));
      float sum = 0.f;
#pragma unroll
      for (int i = 0; i < 16; ++i) {
        float e = __expf(al[(kbase + i) * AS + s] - mx);
        al[(kbase + i) * AS + s] = e;
        sum += e;
      }
      sum += __shfl_xor(sum, 8, 32);
      sum += __shfl_xor(sum, 16, 32);
      const float inv = 1.0f / sum;
#pragma unroll
      for (int i = 0; i < 16; ++i) al[(kbase + i) * AS + s] *= inv;
    }
    __syncthreads();

    // ---- GEMM2: V[64 x 144] += A[64 x 64] * x_aug^T[64 x 144]
    // (x row 128 is ones -> V[:,128] accumulates sum_s a[k,s])
#pragma unroll
    for (int t = 0; t < 5; ++t) {
      const int ti = wave + 8 * t;        // wave-uniform: EXEC stays all-ones
      if (ti < 36) {
        const int mt2 = ti / 9, nt2 = ti - 9 * mt2;
        const int arow = (16 * mt2 + m) * AS;
        const int brow = (16 * nt2 + m) * XS;   // B[s][c'] = x[c'][s]
        v8f acc = vacc[t];
        for (int c = 0; c < SB; c += 4) {
          v2f af;
          af.x = al[arow + c + kb];
          af.y = al[arow + c + kb + 1];
          v2f bf;
          bf.x = xc[brow + c + kb];
          bf.y = xc[brow + c + kb + 1];
          acc = wmma_f32_4(af, bf, acc);
        }
        vacc[t] = acc;
      }
    }
  }

  // ---- Per-workgroup partial [64][144] (no atomics -> deterministic).
  float* P = partial + (size_t)wg * KK * KCOLS;
#pragma unroll
  for (int t = 0; t < 5; ++t) {
    const int ti = wave + 8 * t;
    if (ti < 36) {
      const int mt2 = ti / 9, nt2 = ti - 9 * mt2;
      const int col = 16 * nt2 + m;
#pragma unroll
      for (int r = 0; r < 8; ++r) {
        const int row = 16 * mt2 + r + 8 * (lane >> 4);
        P[row * KCOLS + col] = vacc[t][r];
      }
    }
  }
}

__global__ __launch_bounds__(256)
void netvlad_finalize(const float* __restrict__ partial,    // [N*NSB][64][144]
                      const float* __restrict__ centroids,  // [K][C]
                      float* __restrict__ out)              // [N][K*C]
{
  __shared__ float flatv[KK * CC];
  __shared__ float wsum[8];
  __shared__ float sinv;

  const int n    = blockIdx.x;
  const int tid  = threadIdx.x;
  const int lane = tid & 31;
  const int wave = tid >> 5;
  const float* Pn = partial + (size_t)n * NSB * KK * KCOLS;

  float tot = 0.f;
  for (int i = 0; i < 8; ++i) {
    const int k = wave * 8 + i;
    float sum_a = 0.f;
    float v0 = 0.f, v1 = 0.f, v2 = 0.f, v3 = 0.f;
    for (int sb = 0; sb < NSB; ++sb) {
      const float* Pk = Pn + ((size_t)sb * KK + k) * KCOLS;
      sum_a += Pk[128];
      v0 += Pk[lane];
      v1 += Pk[lane + 32];
      v2 += Pk[lane + 64];
      v3 += Pk[lane + 96];
    }
    const float* cen = centroids + k * CC;
    v0 -= sum_a * cen[lane];
    v1 -= sum_a * cen[lane + 32];
    v2 -= sum_a * cen[lane + 64];
    v3 -= sum_a * cen[lane + 96];
    float sq = v0 * v0 + v1 * v1 + v2 * v2 + v3 * v3;
#pragma unroll
    for (int off = 16; off > 0; off >>= 1) sq += __shfl_xor(sq, off, 32);
    const float inv = 1.0f / fmaxf(sqrtf(sq), 1e-12f);
    v0 *= inv; v1 *= inv; v2 *= inv; v3 *= inv;
    float* row = flatv + k * CC;
    row[lane] = v0; row[lane + 32] = v1; row[lane + 64] = v2; row[lane + 96] = v3;
    tot += v0 * v0 + v1 * v1 + v2 * v2 + v3 * v3;
  }
#pragma unroll
  for (int off = 16; off > 0; off >>= 1) tot += __shfl_xor(tot, off, 32);
  if (lane == 0) wsum[wave] = tot;
  __syncthreads();
  if (tid == 0) {
    float t = 0.f;
    for (int i = 0; i < 8; ++i) t += wsum[i];
    sinv = 1.0f / fmaxf(sqrtf(t), 1e-12f);
  }
  __syncthreads();
  const float invT = sinv;
  float* on = out + (size_t)n * (KK * CC);
  for (int i = tid; i < KK * CC; i += 256) on[i] = flatv[i] * invT;
}

extern "C" void kernel_launch(void* const* d_in, const int* in_sizes, int n_in,
                              void* d_out, int out_size, void* d_ws, size_t ws_size,
                              hipStream_t stream) {
  const float* x         = (const float*)d_in[0];  // [32][128][128][128] f32
  const float* conv_w    = (const float*)d_in[1];  // [64][128] f32
  const float* centroids = (const float*)d_in[2];  // [64][128] f32
  float* out = (float*)d_out;                      // [32][8192] f32
  float* partial = (float*)d_ws;                   // N*NSB*64*144*4 = ~18.9 MB

  (void)in_sizes; (void)n_in; (void)out_size; (void)ws_size;

  netvlad_main<<<NN * NSB, 256, 0, stream>>>(x, conv_w, partial);
  netvlad_finalize<<<NN, 256, 0, stream>>>(partial, centroids, out);
}